// Algebraic_interval_66640712564830
// MI455X (gfx1250) — compile-verified
//
#include <hip/hip_runtime.h>
#include <stdint.h>

typedef __attribute__((ext_vector_type(4))) float    v4f;
typedef __attribute__((ext_vector_type(4))) unsigned v4u;

#define N_COLS   16
#define N_FEAT   696   // 16 + 120 + 560
#define SENT     16    // sentinel index -> LDS slot holding 1.0f

// Compile-time combo table: code = a | b<<5 | c<<10, unused slots = SENT.
// Order matches itertools.combinations: singles, then pairs, then triples,
// each lexicographic.
struct ComboTab {
    unsigned v[N_FEAT];
    constexpr ComboTab() : v{} {
        int n = 0;
        for (int a = 0; a < N_COLS; ++a)
            v[n++] = (unsigned)a | (SENT << 5) | (SENT << 10);
        for (int a = 0; a < N_COLS; ++a)
            for (int b = a + 1; b < N_COLS; ++b)
                v[n++] = (unsigned)a | ((unsigned)b << 5) | (SENT << 10);
        for (int a = 0; a < N_COLS; ++a)
            for (int b = a + 1; b < N_COLS; ++b)
                for (int c = b + 1; c < N_COLS; ++c)
                    v[n++] = (unsigned)a | ((unsigned)b << 5) | ((unsigned)c << 10);
    }
};
__constant__ ComboTab g_tab;

__device__ __forceinline__ float prod3(const float* s, unsigned code) {
    return s[code & 31u] * s[(code >> 5) & 31u] * s[(code >> 10) & 31u];
}

__global__ __launch_bounds__(256) void tnorm_feat_kernel(
    const float* __restrict__ xl, const float* __restrict__ xu,
    float* __restrict__ out, int B)
{
    // per-wave LDS: [0..16] = l-row (+1.0 sentinel), [17..33] = u-row (+1.0), pad to 36
    __shared__ float sx[8][36];
    const int lane = threadIdx.x & 31;
    const int wave = threadIdx.x >> 5;
    const int row  = blockIdx.x * 8 + wave;
    if (row >= B) return;

    float* sl = &sx[wave][0];
    float* su = &sx[wave][17];

    // --- async stage: lanes 0..15 copy xl row, lanes 16..31 copy xu row ---
    {
        const float* gsrc = (lane < 16) ? (xl + (size_t)row * N_COLS + lane)
                                        : (xu + (size_t)row * N_COLS + (lane - 16));
        float* ldst = (lane < 16) ? (sl + lane) : (su + (lane - 16));
        unsigned ldsoff = (unsigned)(uintptr_t)ldst;  // low 32 bits of generic ptr = LDS addr
        asm volatile("global_load_async_to_lds_b32 %0, %1, off"
                     :: "v"(ldsoff), "v"(gsrc) : "memory");
    }
    if (lane == 0) { sl[SENT] = 1.0f; su[SENT] = 1.0f; }  // sentinels (disjoint addrs)
    asm volatile("s_wait_asynccnt 0" ::: "memory");

    float* outl = out + (size_t)row * N_FEAT;
    float* outu = out + (size_t)B * N_FEAT + (size_t)row * N_FEAT;

    #pragma unroll
    for (int i = 0; i < 6; ++i) {
        const int jb = i * 128 + lane * 4;   // 4 consecutive features per lane
        if (jb + 3 < N_FEAT) {               // tail iter: lanes 0..13 only
            const v4u c4 = *(const v4u*)(g_tab.v + jb);
            v4f pl, pu;
            pl.x = prod3(sl, c4.x);  pu.x = prod3(su, c4.x);
            pl.y = prod3(sl, c4.y);  pu.y = prod3(su, c4.y);
            pl.z = prod3(sl, c4.z);  pu.z = prod3(su, c4.z);
            pl.w = prod3(sl, c4.w);  pu.w = prod3(su, c4.w);
            // streaming outputs (365 MB > 192 MB L2): non-temporal 128-bit stores
            __builtin_nontemporal_store(pl, (v4f*)(outl + jb));
            __builtin_nontemporal_store(pu, (v4f*)(outu + jb));
        }
    }
}

extern "C" void kernel_launch(void* const* d_in, const int* in_sizes, int n_in,
                              void* d_out, int out_size, void* d_ws, size_t ws_size,
                              hipStream_t stream) {
    (void)n_in; (void)out_size; (void)d_ws; (void)ws_size;
    const float* xl = (const float*)d_in[0];
    const float* xu = (const float*)d_in[1];
    float* out = (float*)d_out;
    const int B = in_sizes[0] / N_COLS;          // 65536
    const int rowsPerBlock = 8;                   // 8 wave32 per 256-thread block
    dim3 grid((B + rowsPerBlock - 1) / rowsPerBlock);
    dim3 block(256);
    tnorm_feat_kernel<<<grid, block, 0, stream>>>(xl, xu, out, B);
}